// SpatialSingularityRecalibration_81922206204086
// MI455X (gfx1250) — compile-verified
//
#include <hip/hip_runtime.h>
#include <math.h>

typedef __attribute__((ext_vector_type(2))) float v2f;
typedef __attribute__((ext_vector_type(8))) float v8f;

#define Bn   8
#define Hn   64
#define Wn   64
#define Cn   256
#define CRn  32
#define HWCn (Hn * Wn * Cn)          /* 1048576 = 2^20 */
#define NTOT (Bn * HWCn)             /* 8388608 */
#define EPSF 1e-7f
#define BNEPSF 1e-3f
#define CG   8                       /* channels per kplane block */
#define PL   (64 * 65 + 1)           /* padded plane stride (floats); odd -> bank decorrelation across planes */

// ---------------------------------------------------------------------------
// Kernel A1: per-block partial min/max over one 16384-element chunk of batch b
// grid = Bn*64 blocks x 256 threads
// ---------------------------------------------------------------------------
__global__ __launch_bounds__(256) void kminmax_partial(const float* __restrict__ x,
                                                       float* __restrict__ pmin,
                                                       float* __restrict__ pmax) {
    int blk = blockIdx.x;
    int b = blk >> 6;
    int chunk = blk & 63;
    const float4* xb = (const float4*)(x + (size_t)b * HWCn) + (size_t)chunk * 4096;
    float lmin = 3.4e38f, lmax = -3.4e38f;
    for (int i = threadIdx.x; i < 4096; i += 256) {
        float4 v = xb[i];
        lmin = fminf(lmin, fminf(fminf(v.x, v.y), fminf(v.z, v.w)));
        lmax = fmaxf(lmax, fmaxf(fmaxf(v.x, v.y), fmaxf(v.z, v.w)));
    }
    __shared__ float smin[256], smax[256];
    int t = threadIdx.x;
    smin[t] = lmin; smax[t] = lmax;
    __syncthreads();
    for (int s = 128; s > 0; s >>= 1) {
        if (t < s) {
            smin[t] = fminf(smin[t], smin[t + s]);
            smax[t] = fmaxf(smax[t], smax[t + s]);
        }
        __syncthreads();
    }
    if (t == 0) { pmin[blk] = smin[0]; pmax[blk] = smax[0]; }
}

// ---------------------------------------------------------------------------
// Kernel A2: finalize per-batch min and 1/(max-min+eps). grid = Bn x 64
// ---------------------------------------------------------------------------
__global__ __launch_bounds__(64) void kminmax_final(const float* __restrict__ pmin,
                                                    const float* __restrict__ pmax,
                                                    float* __restrict__ bmin,
                                                    float* __restrict__ binv) {
    int b = blockIdx.x;
    __shared__ float smin[64], smax[64];
    int t = threadIdx.x;
    smin[t] = pmin[b * 64 + t];
    smax[t] = pmax[b * 64 + t];
    __syncthreads();
    for (int s = 32; s > 0; s >>= 1) {
        if (t < s) {
            smin[t] = fminf(smin[t], smin[t + s]);
            smax[t] = fmaxf(smax[t], smax[t + s]);
        }
        __syncthreads();
    }
    if (t == 0) {
        bmin[b] = smin[0];
        binv[b] = 1.0f / (smax[0] - smin[0] + EPSF);
    }
}

// ---------------------------------------------------------------------------
// Kernel B: one block per (b, 8-channel group). 8 SATs in dynamic LDS ->
// 4 clamped box sums -> alpha, sim, BN+sigmoid. Writes P=sim*sig to out,
// (1-sim) to q, alpha-means to squeeze. grid = Bn*(Cn/CG)=256 blocks x 256.
// Loads are global_load_b128 (4 ch/lane); stores are 32B-contiguous/pixel.
// ---------------------------------------------------------------------------
__global__ __launch_bounds__(256) void kplane(const float* __restrict__ x,
                                              const float* __restrict__ bmin,
                                              const float* __restrict__ binv,
                                              const float* __restrict__ gamma,
                                              const float* __restrict__ beta,
                                              const float* __restrict__ mmean,
                                              const float* __restrict__ mvar,
                                              float* __restrict__ outP,
                                              float* __restrict__ q,
                                              float* __restrict__ squeeze) {
    extern __shared__ float smem[];
    float* sat  = smem;             // CG * PL floats (8 padded 64x64 SATs)
    float* redf = smem + CG * PL;   // 256 floats
    int blk = blockIdx.x;
    int b  = blk >> 5;              // Cn/CG == 32 groups
    int c0 = (blk & 31) * CG;
    int tid = threadIdx.x;
    float xmn = bmin[b], inv = binv[b];
    const float* xb = x + (size_t)b * HWCn + c0;

    // ---- load + normalize: 4096 pixels x 8 ch, as 2 float4 quads per pixel
    for (int idx = tid; idx < 8192; idx += 256) {
        int p  = idx >> 1;
        int cq = (idx & 1) * 4;
        int h = p >> 6, w = p & 63;
        float4 v = *(const float4*)(xb + (size_t)p * Cn + cq);
        int base = h * 65 + w;
        sat[(cq + 0) * PL + base] = (v.x - xmn) * inv;
        sat[(cq + 1) * PL + base] = (v.y - xmn) * inv;
        sat[(cq + 2) * PL + base] = (v.z - xmn) * inv;
        sat[(cq + 3) * PL + base] = (v.w - xmn) * inv;
    }
    __syncthreads();
    // ---- row-wise inclusive prefix: 512 (ch,row) tasks
    for (int task = tid; task < 512; task += 256) {
        int ch = task >> 6, row = task & 63;
        float* r = sat + ch * PL + row * 65;
        float acc = 0.f;
        for (int w = 0; w < 64; ++w) { acc += r[w]; r[w] = acc; }
    }
    __syncthreads();
    // ---- column-wise inclusive prefix: 512 (ch,col) tasks
    for (int task = tid; task < 512; task += 256) {
        int ch = task >> 6, col = task & 63;
        float* cb = sat + ch * PL + col;
        float acc = 0.f;
        for (int h = 0; h < 64; ++h) { acc += cb[h * 65]; cb[h * 65] = acc; }
    }
    __syncthreads();

    const float LN2 = 0.69314718055994531f;
    const float lrc[4] = {-1.5f * LN2, -0.5f * LN2, 0.5f * LN2, 1.5f * LN2};
    const float S2 = 5.0f * LN2 * LN2;
    const int ks[4] = {2, 4, 8, 16};

    int ch = tid & 7;               // invariant per thread (256 % 8 == 0)
    int c  = c0 + ch;
    const float* plane = sat + ch * PL;
    float bnscale = gamma[c] * rsqrtf(mvar[c] + BNEPSF);
    float bnbias  = beta[c];
    float bnmean  = mmean[c];

    auto SATP = [&](int r, int cc) -> float {
        if (r < 0 || cc < 0) return 0.f;
        r  = r  > 63 ? 63 : r;
        cc = cc > 63 ? 63 : cc;
        return plane[r * 65 + cc];
    };

    float asum = 0.f;
    float* outPb = outP + (size_t)b * HWCn + c;
    float* qb    = q    + (size_t)b * HWCn + c;

    for (int idx = tid; idx < 32768; idx += 256) {
        int p = idx >> 3;           // (idx & 7) == ch
        int h = p >> 6, w = p & 63;
        float lm[4];
#pragma unroll
        for (int s = 0; s < 4; ++s) {
            int k  = ks[s];
            int pb = (k - 1) >> 1;  // SAME pad_lo (matches XLA)
            int r0 = h - pb, r1 = h - pb + k - 1;
            int col0 = w - pb, col1 = w - pb + k - 1;
            float box = SATP(r1, col1) - SATP(r0 - 1, col1)
                      - SATP(r1, col0 - 1) + SATP(r0 - 1, col0 - 1);
            lm[s] = logf(box + EPSF);
        }
        float dot = lm[0] * lrc[0] + lm[1] * lrc[1] + lm[2] * lrc[2] + lm[3] * lrc[3];
        float alpha = dot / S2;
        float mean = 0.25f * (lm[0] + lm[1] + lm[2] + lm[3]);
        float num = 0.f, na2 = 0.f, nb2 = 0.f;
#pragma unroll
        for (int s = 0; s < 4; ++s) {
            float a  = lm[s] - mean;
            float bb = alpha * lrc[s];   // centered OLS estimate = alpha * lr_c
            num += a * bb; na2 += a * a; nb2 += bb * bb;
        }
        float sim = num / (sqrtf(na2) * sqrtf(nb2) + EPSF);
        float an = (alpha - bnmean) * bnscale + bnbias;
        float sg = 1.0f / (1.0f + expf(-an));
        outPb[(size_t)p * Cn] = sim * sg;   // P term
        qb[(size_t)p * Cn]    = 1.0f - sim; // (1 - sim)
        asum += alpha;
    }
    redf[tid] = asum;
    __syncthreads();
    for (int s = 128; s >= 8; s >>= 1) {    // stride >= 8 preserves channel
        if (tid < s) redf[tid] += redf[tid + s];
        __syncthreads();
    }
    if (tid < 8) squeeze[b * Cn + c0 + tid] = redf[tid] * (1.0f / 4096.0f);
}

// ---------------------------------------------------------------------------
// Kernel C: SE MLP via V_WMMA_F32_16X16X4_F32. One wave (32 threads).
// A fragments come from LDS with a 0/1 lane mask -> no exec-mask branching.
// GEMM1: [16x256]x[256x32] (M=8 valid). GEMM2: [16x32]x[32x256].
// ---------------------------------------------------------------------------
__global__ __launch_bounds__(32) void kse(const float* __restrict__ squeeze,
                                          const float* __restrict__ w1,
                                          const float* __restrict__ b1,
                                          const float* __restrict__ w2,
                                          const float* __restrict__ b2,
                                          float* __restrict__ excite) {
    __shared__ float sq[Bn * Cn];    // 2048 floats
    __shared__ float h1[8][CRn];
    int lane = threadIdx.x;
    for (int i = lane; i < (Bn * Cn) / 4; i += 32)
        ((float4*)sq)[i] = ((const float4*)squeeze)[i];
    __syncthreads();

    int half = lane >> 4;            // 0: K=0,1 | 1: K=2,3 (ISA 16x4 f32 A layout)
    int mn = lane & 15;
    int ms = mn & 7;                 // clamped row index (rows 8..15 are zero)
    float amask = (mn < 8) ? 1.0f : 0.0f;

    // ---- GEMM1 + bias + relu -> h1[8][32]
    for (int nt = 0; nt < 2; ++nt) {
        v8f acc = {};
        for (int kk = 0; kk < Cn; kk += 4) {
            int kb = kk + half * 2;
            v2f a, bf;
            a.x = sq[ms * Cn + kb] * amask;
            a.y = sq[ms * Cn + kb + 1] * amask;
            int n = nt * 16 + mn;
            bf.x = w1[kb * CRn + n];
            bf.y = w1[(kb + 1) * CRn + n];
            acc = __builtin_amdgcn_wmma_f32_16x16x4_f32(
                false, a, false, bf, (short)0, acc, false, false);
        }
        if (lane < 16) {             // lanes 0..15 hold M = vgpr index (0..7)
            int n = nt * 16 + lane;
#pragma unroll
            for (int j = 0; j < 8; ++j) {
                float v = acc[j] + b1[n];
                h1[j][n] = v > 0.f ? v : 0.f;
            }
        }
        __syncthreads();
    }

    // ---- GEMM2 + bias + sigmoid -> excite[8][256]
    for (int nt = 0; nt < 16; ++nt) {
        v8f acc = {};
#pragma unroll
        for (int kk = 0; kk < CRn; kk += 4) {
            int kb = kk + half * 2;
            v2f a, bf;
            a.x = h1[ms][kb] * amask;
            a.y = h1[ms][kb + 1] * amask;
            int n = nt * 16 + mn;
            bf.x = w2[kb * Cn + n];
            bf.y = w2[(kb + 1) * Cn + n];
            acc = __builtin_amdgcn_wmma_f32_16x16x4_f32(
                false, a, false, bf, (short)0, acc, false, false);
        }
        if (lane < 16) {
            int n = nt * 16 + lane;
#pragma unroll
            for (int j = 0; j < 8; ++j) {
                float e = acc[j] + b2[n];
                excite[j * Cn + n] = 1.0f / (1.0f + expf(-e));
            }
        }
        __syncthreads();
    }
}

// ---------------------------------------------------------------------------
// Kernel D: out = P + (1-sim) * x_norm * excite[b,c]   (float4 vectorized)
// grid = NTOT/4/256 blocks x 256 threads
// ---------------------------------------------------------------------------
__global__ __launch_bounds__(256) void kcombine(const float* __restrict__ x,
                                                const float* __restrict__ q,
                                                const float* __restrict__ excite,
                                                const float* __restrict__ bmin,
                                                const float* __restrict__ binv,
                                                float* __restrict__ out) {
    size_t i4 = (size_t)blockIdx.x * blockDim.x + threadIdx.x;
    size_t i = i4 * 4;
    int b = (int)(i >> 20);          // HWC = 2^20
    int c = (int)(i & (Cn - 1));
    float xmn = bmin[b], inv = binv[b];
    float4 xv = ((const float4*)x)[i4];
    float4 qv = ((const float4*)q)[i4];
    float4 pv = ((const float4*)out)[i4];
    float4 ev = *(const float4*)(excite + b * Cn + c);
    float4 r;
    r.x = pv.x + qv.x * ((xv.x - xmn) * inv) * ev.x;
    r.y = pv.y + qv.y * ((xv.y - xmn) * inv) * ev.y;
    r.z = pv.z + qv.z * ((xv.z - xmn) * inv) * ev.z;
    r.w = pv.w + qv.w * ((xv.w - xmn) * inv) * ev.w;
    ((float4*)out)[i4] = r;
}

// ---------------------------------------------------------------------------
extern "C" void kernel_launch(void* const* d_in, const int* in_sizes, int n_in,
                              void* d_out, int out_size, void* d_ws, size_t ws_size,
                              hipStream_t stream) {
    const float* x     = (const float*)d_in[0];
    const float* w1    = (const float*)d_in[1];
    const float* b1    = (const float*)d_in[2];
    const float* w2    = (const float*)d_in[3];
    const float* b2    = (const float*)d_in[4];
    const float* gamma = (const float*)d_in[5];
    const float* beta  = (const float*)d_in[6];
    const float* mmean = (const float*)d_in[7];
    const float* mvar  = (const float*)d_in[8];
    float* out = (float*)d_out;

    float* ws      = (float*)d_ws;
    float* q       = ws;                     // NTOT floats: (1-sim)
    float* squeeze = q + NTOT;               // Bn*Cn
    float* excite  = squeeze + Bn * Cn;      // Bn*Cn
    float* pmin    = excite + Bn * Cn;       // Bn*64
    float* pmax    = pmin + Bn * 64;         // Bn*64
    float* bmn     = pmax + Bn * 64;         // Bn
    float* binv    = bmn + Bn;               // Bn

    size_t lds_bytes = (size_t)(CG * PL + 256) * sizeof(float);  // ~134 KB of 320 KB WGP LDS

    kminmax_partial<<<Bn * 64, 256, 0, stream>>>(x, pmin, pmax);
    kminmax_final<<<Bn, 64, 0, stream>>>(pmin, pmax, bmn, binv);
    kplane<<<Bn * (Cn / CG), 256, lds_bytes, stream>>>(x, bmn, binv, gamma, beta,
                                                       mmean, mvar, out, q, squeeze);
    kse<<<1, 32, 0, stream>>>(squeeze, w1, b1, w2, b2, excite);
    kcombine<<<NTOT / 4 / 256, 256, 0, stream>>>(x, q, excite, bmn, binv, out);
}